// EmbeddingsLayer_72782515798476
// MI455X (gfx1250) — compile-verified
//
#include <hip/hip_runtime.h>
#include <hip/hip_bf16.h>
#include <stdint.h>

// Problem constants (match the reference).
#define B_  4096
#define S_  512
#define D_  18
#define WPB 8          // waves per block (wave32) -> 256 threads, one batch row per wave
#define EPS_ 1e-7f

// ---- CDNA5 async global->LDS copies (ASYNCcnt path, ISA 15.18.3 op 96/98) ----
__device__ __forceinline__ void async_g2l_b128(uint32_t lds_addr, const void* gaddr) {
  asm volatile("global_load_async_to_lds_b128 %0, %1, off"
               :: "v"(lds_addr), "v"(gaddr) : "memory");
}
__device__ __forceinline__ void async_g2l_b32(uint32_t lds_addr, const void* gaddr) {
  asm volatile("global_load_async_to_lds_b32 %0, %1, off"
               :: "v"(lds_addr), "v"(gaddr) : "memory");
}
__device__ __forceinline__ void wait_async0() {
  asm volatile("s_wait_asynccnt 0" ::: "memory");
}

__global__ __launch_bounds__(32 * WPB)
void emb_metric_kernel(const int* __restrict__ mids,
                       const int* __restrict__ wids,
                       const int* __restrict__ cnt,
                       const float* __restrict__ mc,
                       const float* __restrict__ xl_t,
                       const float* __restrict__ xr_t,
                       float* __restrict__ out)
{
  // Per-wave private LDS regions -> no inter-wave sharing -> no barriers.
  __shared__ __align__(16) int   sW[WPB][S_];
  __shared__ __align__(16) int   sM[WPB][S_];
  __shared__ __align__(16) float sMC[WPB][D_ * D_];

  const int lane = threadIdx.x & 31;
  const int wave = threadIdx.x >> 5;
  const int b    = blockIdx.x * WPB + wave;

  const int scnt = cnt[b];                 // uniform within the wave (scalar)

  // ---- Stage index rows + mc_table into LDS asynchronously ----
  const int* gw = wids + (size_t)b * S_;
  const int* gm = mids + (size_t)b * S_;
  const uint32_t lw  = (uint32_t)(uintptr_t)&sW[wave][0];
  const uint32_t lm  = (uint32_t)(uintptr_t)&sM[wave][0];
  const uint32_t lmc = (uint32_t)(uintptr_t)&sMC[wave][0];

  // 512 dwords per row; round r covers dwords [128r, 128r+128). Only copy
  // rounds that can be touched (indices 0..scnt-1 are ever read).
#pragma unroll
  for (int r = 0; r < 4; ++r) {
    if (r * 128 < scnt) {
      const int di = r * 128 + lane * 4;   // 16B-aligned chunks
      async_g2l_b128(lw + (uint32_t)di * 4u, gw + di);
      async_g2l_b128(lm + (uint32_t)di * 4u, gm + di);
    }
  }
  // mc_table: 18*18 = 324 dwords.
  for (int i = lane; i < D_ * D_; i += 32)
    async_g2l_b32(lmc + (uint32_t)i * 4u, mc + i);

  wait_async0();                           // per-wave ASYNCcnt -> LDS is ready

  // ---- head = xl_table[wids[b,0]] ----
  const int wid0 = sW[wave][0];
  if (lane < D_)                           // second output: head, (B,18) at offset B*18
    out[(size_t)B_ * D_ + (size_t)b * D_ + lane] = xl_t[(size_t)wid0 * D_ + lane];

  float h[D_];
  {
    const float2* hp = (const float2*)(xl_t + (size_t)wid0 * D_);  // 8B aligned (72B rows)
#pragma unroll
    for (int k = 0; k < 9; ++k) { float2 t = hp[k]; h[2*k] = t.x; h[2*k+1] = t.y; }
  }

  // ---- Per-lane accumulation over a contiguous chunk of positions ----
  float pv[D_], vs[D_], hs[D_];
#pragma unroll
  for (int d = 0; d < D_; ++d) { pv[d] = 0.f; vs[d] = 0.f; hs[d] = 0.f; }

  const int npos = scnt - 1;               // positions 1..scnt-1
  const int L    = (npos + 31) >> 5;
  const int p0   = 1 + lane * L;
  const int pe   = (p0 + L < scnt) ? (p0 + L) : scnt;

  if (p0 < pe) {
    { // first position of the chunk: h-term + seed pv = mb*xr (no v-term yet)
      const int wid = sW[wave][p0], mid = sM[wave][p0];
      const float2* xl2 = (const float2*)(xl_t + (size_t)wid * D_);
      const float2* xr2 = (const float2*)(xr_t + (size_t)wid * D_);
      const float2* mb2 = (const float2*)(&sMC[wave][mid * D_]);
#pragma unroll
      for (int k = 0; k < 9; ++k) {
        const float2 mb = mb2[k], xl = xl2[k], xr = xr2[k];
        const int d = 2 * k;
        const float hx = mb.x * (0.5f * (xl.x + xr.x) - h[d]);
        const float hy = mb.y * (0.5f * (xl.y + xr.y) - h[d+1]);
        hs[d]   += hx * hx;  hs[d+1] += hy * hy;
        pv[d]    = mb.x * xr.x;  pv[d+1] = mb.y * xr.y;
      }
    }
    // steady state: each new row gives v-term (vs prev product) + h-term
    for (int s = p0 + 1; s < pe; ++s) {
      const int wid = sW[wave][s], mid = sM[wave][s];
      const float2* xl2 = (const float2*)(xl_t + (size_t)wid * D_);
      const float2* xr2 = (const float2*)(xr_t + (size_t)wid * D_);
      const float2* mb2 = (const float2*)(&sMC[wave][mid * D_]);
#pragma unroll
      for (int k = 0; k < 9; ++k) {
        const float2 mb = mb2[k], xl = xl2[k], xr = xr2[k];
        const int d = 2 * k;
        const float vx = pv[d]   - mb.x * xl.x;
        const float vy = pv[d+1] - mb.y * xl.y;
        vs[d]   += vx * vx;  vs[d+1] += vy * vy;
        const float hx = mb.x * (0.5f * (xl.x + xr.x) - h[d]);
        const float hy = mb.y * (0.5f * (xl.y + xr.y) - h[d+1]);
        hs[d]   += hx * hx;  hs[d+1] += hy * hy;
        pv[d]    = mb.x * xr.x;  pv[d+1] = mb.y * xr.y;
      }
    }
    // chunk boundary: v-term j = pe-1 needs row pe (xl + mb only)
    if (pe < scnt) {
      const int wid = sW[wave][pe], mid = sM[wave][pe];
      const float2* xl2 = (const float2*)(xl_t + (size_t)wid * D_);
      const float2* mb2 = (const float2*)(&sMC[wave][mid * D_]);
#pragma unroll
      for (int k = 0; k < 9; ++k) {
        const float2 mb = mb2[k], xl = xl2[k];
        const int d = 2 * k;
        const float vx = pv[d]   - mb.x * xl.x;
        const float vy = pv[d+1] - mb.y * xl.y;
        vs[d] += vx * vx;  vs[d+1] += vy * vy;
      }
    }
  }

  // ---- wave32 butterfly reduction of the 2x18 partials ----
#pragma unroll
  for (int m = 1; m < 32; m <<= 1) {
#pragma unroll
    for (int d = 0; d < D_; ++d) {
      vs[d] += __shfl_xor(vs[d], m, 32);
      hs[d] += __shfl_xor(hs[d], m, 32);
    }
  }

  if (lane == 0) {
    const float inv = 1.0f / ((float)scnt + EPS_);
    float2* o2 = (float2*)(out + (size_t)b * D_);   // 8B aligned (b*72)
#pragma unroll
    for (int k = 0; k < 9; ++k) {
      float2 t;
      t.x = 0.5f * (__builtin_sqrtf(vs[2*k]   * inv) + __builtin_sqrtf(hs[2*k]   * inv));
      t.y = 0.5f * (__builtin_sqrtf(vs[2*k+1] * inv) + __builtin_sqrtf(hs[2*k+1] * inv));
      o2[k] = t;
    }
  }
}

extern "C" void kernel_launch(void* const* d_in, const int* in_sizes, int n_in,
                              void* d_out, int out_size, void* d_ws, size_t ws_size,
                              hipStream_t stream) {
  (void)in_sizes; (void)n_in; (void)out_size; (void)d_ws; (void)ws_size;
  const int*   mids = (const int*)d_in[0];
  const int*   wids = (const int*)d_in[1];
  const int*   cntp = (const int*)d_in[2];
  const float* mc   = (const float*)d_in[3];
  const float* xl   = (const float*)d_in[4];
  const float* xr   = (const float*)d_in[5];
  float* out = (float*)d_out;

  dim3 grid(B_ / WPB);
  dim3 block(32 * WPB);
  hipLaunchKernelGGL(emb_metric_kernel, grid, block, 0, stream,
                     mids, wids, cntp, mc, xl, xr, out);
}